// SimplificationLoss_28621662060808
// MI455X (gfx1250) — compile-verified
//
#include <hip/hip_runtime.h>
#include <hip/hip_bf16.h>
#include <math.h>

// Chamfer distance on MI455X (gfx1250, wave32).
// d2(i,j) = ||p_i||^2 - 2 p_i.q_j + ||q_j||^2 computed as a K=4 WMMA:
//   A row i  = [p0, p1, p2, ||p||^2]        (16x4 fp32)
//   B col j  = [-2q0, -2q1, -2q2, 1]        (4x16 fp32)
//   D = A*B  = ||p||^2 - 2 p.q   (16x16 fp32 accum, 8 VGPRs/lane)
// then + ||q_j||^2 per column, clamp >= 0, running min over j tiles.

typedef float v2f __attribute__((ext_vector_type(2)));
typedef float v8f __attribute__((ext_vector_type(8)));

// out[b,i] = min_j max(0, ||P[b,i,:] - Q[b,j,:]||^2)
// P: [B, Np, 3], Q: [B, Nq, 3], out: [B, Np]
// Block = 256 threads = 8 waves; each wave owns 16 rows; Np % 128 == 0, Nq % 16 == 0.
__global__ __launch_bounds__(256) void chamfer_min_kernel(
    const float* __restrict__ P, const float* __restrict__ Q,
    float* __restrict__ out, int Np, int Nq)
{
    const int tilesPerBatch = Np >> 7;           // Np / 128
    const int b    = blockIdx.x / tilesPerBatch;
    const int tile = blockIdx.x % tilesPerBatch;
    const int lane = threadIdx.x & 31;
    const int wv   = threadIdx.x >> 5;
    const int rowbase = tile * 128 + wv * 16;
    const int half = lane >> 4;                  // 0 -> K={0,1}, 1 -> K={2,3}
    const int lc   = lane & 15;                  // matrix column / row-in-tile owner

    // ---- A fragment (constant across the j loop) ----
    // ISA 16x4 A layout: lane l holds A[l&15][2*(l>>4) + {0,1}]
    const float* prow = P + ((size_t)b * Np + (rowbase + lc)) * 3;
    const float p0 = prow[0], p1 = prow[1], p2 = prow[2];
    const float s2 = p0 * p0 + p1 * p1 + p2 * p2;
    v2f afrag;
    afrag.x = half ? p2 : p0;
    afrag.y = half ? s2 : p1;

    v8f rmin;
#pragma unroll
    for (int v = 0; v < 8; ++v) rmin[v] = 3.4028235e38f;

    const float* Qb = Q + (size_t)b * Nq * 3;
    for (int jb = 0; jb < Nq; jb += 16) {
        // B fragment: lane l holds B[2*(l>>4) + {0,1}][l&15]; column point q = Q[b][jb+lc]
        const float* qrow = Qb + (size_t)(jb + lc) * 3;
        const float q0 = qrow[0], q1 = qrow[1], q2 = qrow[2];
        const float r2 = q0 * q0 + q1 * q1 + q2 * q2;
        v2f bfrag;
        bfrag.x = half ? (-2.0f * q2) : (-2.0f * q0);
        bfrag.y = half ? 1.0f         : (-2.0f * q1);

        v8f c = {0.f, 0.f, 0.f, 0.f, 0.f, 0.f, 0.f, 0.f};
        // (neg_a, A, neg_b, B, c_mod, C, reuse_a, reuse_b)
        c = __builtin_amdgcn_wmma_f32_16x16x4_f32(
                false, afrag, false, bfrag, (short)0, c, false, false);

#pragma unroll
        for (int v = 0; v < 8; ++v) {
            float d2 = c[v] + r2;        // + ||q||^2 (lane owns one column)
            d2 = fmaxf(d2, 0.0f);        // numerical floor, matches reference
            rmin[v] = fminf(rmin[v], d2);
        }
    }

    // Cross-lane min over the 16 columns (xor masks 1,2,4,8 stay inside each
    // 16-lane half of the wave32; lanes 0-15 hold rows 0-7, lanes 16-31 rows 8-15).
#pragma unroll
    for (int m = 8; m >= 1; m >>= 1) {
#pragma unroll
        for (int v = 0; v < 8; ++v)
            rmin[v] = fminf(rmin[v], __shfl_xor(rmin[v], m, 32));
    }

    if (lc == 0) {
        float* o = out + (size_t)b * Np + rowbase + half * 8;
#pragma unroll
        for (int v = 0; v < 8; ++v) o[v] = rmin[v];
    }
}

// One block per batch: cd_p[b] = 0.5*(mean(sqrt(c1)) + mean(sqrt(c2)))
//                      cd_t[b] = mean(c1) + mean(c2)
__global__ __launch_bounds__(256) void chamfer_reduce_kernel(
    const float* __restrict__ c1, const float* __restrict__ c2,
    float* __restrict__ outp, int N, int M, int B)
{
    const int b = blockIdx.x;
    float s1 = 0.f, s1s = 0.f, s2 = 0.f, s2s = 0.f;
    for (int i = threadIdx.x; i < N; i += blockDim.x) {
        float v = c1[(size_t)b * N + i];
        s1 += v; s1s += sqrtf(v);
    }
    for (int i = threadIdx.x; i < M; i += blockDim.x) {
        float v = c2[(size_t)b * M + i];
        s2 += v; s2s += sqrtf(v);
    }
    __shared__ float sh[4][256];
    sh[0][threadIdx.x] = s1;  sh[1][threadIdx.x] = s1s;
    sh[2][threadIdx.x] = s2;  sh[3][threadIdx.x] = s2s;
    __syncthreads();
    for (int off = 128; off > 0; off >>= 1) {
        if ((int)threadIdx.x < off) {
#pragma unroll
            for (int k = 0; k < 4; ++k)
                sh[k][threadIdx.x] += sh[k][threadIdx.x + off];
        }
        __syncthreads();
    }
    if (threadIdx.x == 0) {
        const float m1  = sh[0][0] / (float)N;
        const float m1s = sh[1][0] / (float)N;
        const float m2  = sh[2][0] / (float)M;
        const float m2s = sh[3][0] / (float)M;
        outp[b]     = 0.5f * (m1s + m2s);   // cd_p
        outp[B + b] = m1 + m2;              // cd_t
    }
}

extern "C" void kernel_launch(void* const* d_in, const int* in_sizes, int n_in,
                              void* d_out, int out_size, void* d_ws, size_t ws_size,
                              hipStream_t stream) {
    // setup_inputs order: ref_pc [B,M,3], samp_pc [B,N,3]
    const float* ref  = (const float*)d_in[0];
    const float* samp = (const float*)d_in[1];
    float* out = (float*)d_out;

    const int B = out_size / 2;              // tuple (cd_p[B], cd_t[B]) -> 32 floats
    const int M = in_sizes[0] / (3 * B);     // 8192
    const int N = in_sizes[1] / (3 * B);     // 2048

    // Scratch: c1 = cost_p1_p2 [B,N], c2 = cost_p2_p1 [B,M]  (640 KB total)
    float* c1 = (float*)d_ws;
    float* c2 = c1 + (size_t)B * N;

    // Direction 1: for each samp point, min over ref points.
    chamfer_min_kernel<<<B * (N / 128), 256, 0, stream>>>(samp, ref, c1, N, M);
    // Direction 2: for each ref point, min over samp points (roles swapped).
    chamfer_min_kernel<<<B * (M / 128), 256, 0, stream>>>(ref, samp, c2, M, N);
    // Final per-batch means -> (cd_p, cd_t)
    chamfer_reduce_kernel<<<B, 256, 0, stream>>>(c1, c2, out, N, M, B);
}